// SpikingNeuralNetwork_46763603918992
// MI455X (gfx1250) — compile-verified
//
#include <hip/hip_runtime.h>

typedef float v2f __attribute__((ext_vector_type(2)));
typedef float v8f __attribute__((ext_vector_type(8)));
typedef unsigned int u32x4 __attribute__((ext_vector_type(4)));
typedef int i32x8 __attribute__((ext_vector_type(8)));
typedef int i32x4 __attribute__((ext_vector_type(4)));

#define TCMAIN  16          // timesteps per main chunk
#define NCH     32          // 31 chunks of 16 + 1 chunk of 4  (= 500)
#define NW      8           // waves per block (2 per SIMD for latency hiding)

// LDS float-index layout (dynamic shared, 270336 bytes total).
// X tiles FIRST so Phase-A ds_load offsets (kc*4352 B <= 65280) fit the 16-bit DS immediate.
#define XBASE   0           // x tiles, raw [f64][b16][t16 + 1 pad]       -> 2 * 17408 floats
#define XSTRIDE 17408
#define XROW    17          // padded row stride in dwords (16 t + 1 pad)
#define Z1BASE  34816       // z1 A-layout:  [tc][kc16][lane32][2]        -> 16384 floats
#define BUFB    51200       // cur1 [tc][nt4][r8][ln32] / cur2+z2 / cur3  -> 16384 floats

static __device__ __forceinline__ v8f wmma_f32(v2f a, v2f b, v8f c) {
  // D = A(16x4) * B(4x16) + C, fp32
  return __builtin_amdgcn_wmma_f32_16x16x4_f32(false, a, false, b, (short)0, c, false, false);
}

// Issue one TDM DMA: x[b0:b0+16, 0:64, t0:t0+tlen] -> LDS raw tile with row padding.
static __device__ __forceinline__ void tdm_issue(const float* xg, unsigned lds_byte,
                                                 int b0, int t0, int tlen,
                                                 unsigned pad_iv, unsigned pad_amt) {
  unsigned long long ga = (unsigned long long)(const void*)xg
                        + ((unsigned long long)b0 * 64ull * 500ull + (unsigned long long)t0) * 4ull;
  u32x4 g0;
  g0[0] = 1u;                                            // count=1 (valid), user mode, no gather
  g0[1] = lds_byte;                                      // lds_addr [63:32]
  g0[2] = (unsigned)(ga & 0xFFFFFFFFu);                  // global_addr low   [95:64]
  g0[3] = (unsigned)((ga >> 32) & 0x1FFFFFFu) | (2u << 30); // addr hi [120:96] | type=2 [127:126]

  i32x8 g1;
  g1[0] = (int)((2u << 16) | (1u << 20) | (pad_iv << 22) | (pad_amt << 25)); // 4B elems, pad_enable
  g1[1] = (int)(500u << 16);            // tensor_dim0 = 500 (low16 at [63:48])
  g1[2] = (int)(64u << 16);             // tensor_dim0 hi | tensor_dim1 = 64
  g1[3] = (int)((unsigned)tlen << 16);  // tensor_dim1 hi | tile_dim0 = tlen (t)
  g1[4] = (int)(64u | (16u << 16));     // tile_dim1 = 64 (f) | tile_dim2 = 16 (b)
  g1[5] = 500;                          // tensor_dim0_stride = 500 elements (f lines)
  g1[6] = (int)(32000u << 16);          // stride0 hi | tensor_dim1_stride = 32000 (b lines)
  g1[7] = 0;

  i32x4 g2;
  g2[0] = 2048;                         // tensor_dim2 (b)
  g2[1] = 0;                            // tensor_dim3 (unused), iterate off
  g2[2] = 32000;                        // tensor_dim2_stride low32
  g2[3] = 0;                            // stride hi | tile_dim3 = 0
  i32x4 g3; g3[0] = 0; g3[1] = 0; g3[2] = 0; g3[3] = 0;
  i32x8 gz; gz[0] = 0; gz[1] = 0; gz[2] = 0; gz[3] = 0; gz[4] = 0; gz[5] = 0; gz[6] = 0; gz[7] = 0;

  // clang-23 / therock lane: 6-arg form (u32x4, i32x8, i32x4, i32x4, i32x8, i32 cpol)
  __builtin_amdgcn_tensor_load_to_lds(g0, g1, g2, g3, gz, 0);
}

__global__ __launch_bounds__(256, 1)
void snn_lif3_fused(const float* __restrict__ xg,
                    const float* __restrict__ w1g, const float* __restrict__ b1g,
                    const float* __restrict__ w2g, const float* __restrict__ b2g,
                    const float* __restrict__ w3g, const float* __restrict__ b3g,
                    float* __restrict__ out)
{
  extern __shared__ float S[];

  const int tid  = threadIdx.x;
  const int w    = tid >> 5;
  const int lane = tid & 31;
  const int lo   = lane & 15;
  const int hi   = lane >> 4;
  const int b0   = blockIdx.x * 16;

  // Persistent LIF states (fp32, WMMA C/D layout). v' = 0.9v + 0.1i ; i' = 0.8i + cur ; spike at v' > 1.
  v8f v1 = {}, i1 = {};            // layer1: waves 0-3 own hidden tile [16w,16w+16)
  v8f v2s = {}, i2s = {};          // layer2: waves 0,1 own tiles
  v8f v3 = {}, i3 = {}, acc3 = {}; // layer3: wave 0 (cols 0,1 valid)

  // Prologue: kick off TDM for chunk 0 into x-tile buffer 0.
  if (w == 0) {
    unsigned lb0 = (unsigned)(size_t)(const void*)(S + XBASE);
    tdm_issue(xg, lb0, b0, 0, TCMAIN, 3u /*16 dwords*/, 0u /*1 dword pad*/);
  }

  for (int c = 0; c < NCH; ++c) {
    const int tlen = (c == NCH - 1) ? 4 : TCMAIN;

    // ---- async pipeline: issue chunk c+1, then wait until chunk c's tile is resident ----
    if (w == 0) {
      if (c + 1 < NCH) {
        int tn = (c + 1 == NCH - 1) ? 4 : TCMAIN;
        unsigned piv = (c + 1 == NCH - 1) ? 1u : 3u;   // 4-dword vs 16-dword interval
        unsigned pam = (c + 1 == NCH - 1) ? 12u : 0u;  // 13 vs 1 pad dwords -> stride 17 both
        unsigned lb = (unsigned)(size_t)(const void*)(S + XBASE + ((c + 1) & 1) * XSTRIDE);
        tdm_issue(xg, lb, b0, (c + 1) * TCMAIN, tn, piv, pam);
        __builtin_amdgcn_s_wait_tensorcnt(1);   // chunk c complete, c+1 may still fly
      } else {
        __builtin_amdgcn_s_wait_tensorcnt(0);
      }
    }
    __syncthreads();

    const float* xb = S + XBASE + (c & 1) * XSTRIDE;   // raw [f][b][t+pad] tile

    // ---- Phase A: cur1[t] = x_t @ W1^T + b1 (waves split timesteps) ----
    #pragma unroll
    for (int nt = 0; nt < 4; ++nt) {
      v2f wb[16];
      #pragma unroll
      for (int kc = 0; kc < 16; ++kc)
        wb[kc] = *(const v2f*)(w1g + (nt * 16 + lo) * 64 + kc * 4 + hi * 2);
      float bias = b1g[nt * 16 + lo];
      for (int tc = w; tc < tlen; tc += NW) {
        // batch-preload all 16 A-operand pairs (immediate DS offsets, deep pipeline)
        const float* p0 = xb + tc + hi * 544 + lo * XROW;   // f = kc*4 + hi*2
        const float* p1 = p0 + 16 * XROW;                   // f + 1
        v2f a[16];
        #pragma unroll
        for (int kc = 0; kc < 16; ++kc) {
          a[kc][0] = p0[kc * 1088];
          a[kc][1] = p1[kc * 1088];
        }
        v8f acc;
        #pragma unroll
        for (int r = 0; r < 8; ++r) acc[r] = bias;
        #pragma unroll
        for (int kc = 0; kc < 16; ++kc) acc = wmma_f32(a[kc], wb[kc], acc);
        float* q = S + BUFB + (tc * 4 + nt) * 256 + lane;   // cur1
        #pragma unroll
        for (int r = 0; r < 8; ++r) q[r * 32] = acc[r];
      }
    }
    __syncthreads();

    // ---- Phase B: LIF layer 1 (sequential in t, parallel over hidden tiles); emit z1 A-layout ----
    if (w < 4) {
      const int k = (w << 4) + lo, kc = k >> 2, sub = k & 3;
      const int zcol = ((kc << 5) + ((sub >> 1) << 4) + (hi << 3)) * 2 + (sub & 1);
      for (int tc = 0; tc < tlen; ++tc) {
        const float* p = S + BUFB + (tc * 4 + w) * 256 + lane;
        float* zq = S + Z1BASE + tc * 1024 + zcol;
        #pragma unroll
        for (int r = 0; r < 8; ++r) {
          float cur = p[r * 32];
          float vd = 0.9f * v1[r] + 0.1f * i1[r];
          float id = 0.8f * i1[r];
          float z = (vd > 1.0f) ? 1.0f : 0.0f;
          v1[r] = (vd > 1.0f) ? 0.0f : vd;
          i1[r] = id + cur;
          zq[r * 2] = z;
        }
      }
    }
    __syncthreads();

    // ---- Phase C: cur2[t] = z1 @ W2^T + b2 (waves split timesteps) ----
    #pragma unroll
    for (int nt = 0; nt < 2; ++nt) {
      v2f wb[16];
      #pragma unroll
      for (int kc = 0; kc < 16; ++kc)
        wb[kc] = *(const v2f*)(w2g + (nt * 16 + lo) * 64 + kc * 4 + hi * 2);
      float bias = b2g[nt * 16 + lo];
      for (int tc = w; tc < tlen; tc += NW) {
        const float* p = S + Z1BASE + tc * 1024 + lane * 2;
        v2f a[16];
        #pragma unroll
        for (int kc = 0; kc < 16; ++kc) a[kc] = *(const v2f*)(p + kc * 64);
        v8f acc;
        #pragma unroll
        for (int r = 0; r < 8; ++r) acc[r] = bias;
        #pragma unroll
        for (int kc = 0; kc < 16; ++kc) acc = wmma_f32(a[kc], wb[kc], acc);
        float* q = S + BUFB + (tc * 2 + nt) * 256 + lane;   // cur2 (cur1 dead)
        #pragma unroll
        for (int r = 0; r < 8; ++r) q[r * 32] = acc[r];
      }
    }
    __syncthreads();

    // ---- Phase D: LIF layer 2 (waves 0,1); emit z2 A-layout at BUFB+8192 ----
    if (w < 2) {
      const int k = (w << 4) + lo, kc = k >> 2, sub = k & 3;
      const int zcol = ((kc << 5) + ((sub >> 1) << 4) + (hi << 3)) * 2 + (sub & 1);
      for (int tc = 0; tc < tlen; ++tc) {
        const float* p = S + BUFB + (tc * 2 + w) * 256 + lane;
        float* zq = S + BUFB + 8192 + tc * 512 + zcol;
        #pragma unroll
        for (int r = 0; r < 8; ++r) {
          float cur = p[r * 32];
          float vd = 0.9f * v2s[r] + 0.1f * i2s[r];
          float id = 0.8f * i2s[r];
          float z = (vd > 1.0f) ? 1.0f : 0.0f;
          v2s[r] = (vd > 1.0f) ? 0.0f : vd;
          i2s[r] = id + cur;
          zq[r * 2] = z;
        }
      }
    }
    __syncthreads();

    // ---- Phase E: cur3[t] = z2 @ W3^T + b3 (K=32, N padded to 16; cols 0,1 valid) ----
    {
      v2f wb[8];
      #pragma unroll
      for (int kc = 0; kc < 8; ++kc) {
        if (lo < 2) wb[kc] = *(const v2f*)(w3g + lo * 32 + kc * 4 + hi * 2);
        else { wb[kc][0] = 0.0f; wb[kc][1] = 0.0f; }
      }
      float bias = (lo < 2) ? b3g[lo] : 0.0f;
      for (int tc = w; tc < tlen; tc += NW) {
        const float* p = S + BUFB + 8192 + tc * 512 + lane * 2;
        v2f a[8];
        #pragma unroll
        for (int kc = 0; kc < 8; ++kc) a[kc] = *(const v2f*)(p + kc * 64);
        v8f acc;
        #pragma unroll
        for (int r = 0; r < 8; ++r) acc[r] = bias;
        #pragma unroll
        for (int kc = 0; kc < 8; ++kc) acc = wmma_f32(a[kc], wb[kc], acc);
        float* q = S + BUFB + tc * 256 + lane;              // cur3 (cur2 dead)
        #pragma unroll
        for (int r = 0; r < 8; ++r) q[r * 32] = acc[r];
      }
    }
    __syncthreads();

    // ---- Phase F: LIF layer 3 + spike accumulation (wave 0) ----
    if (w == 0) {
      for (int tc = 0; tc < tlen; ++tc) {
        const float* p = S + BUFB + tc * 256 + lane;
        #pragma unroll
        for (int r = 0; r < 8; ++r) {
          float cur = p[r * 32];
          float vd = 0.9f * v3[r] + 0.1f * i3[r];
          float id = 0.8f * i3[r];
          float z = (vd > 1.0f) ? 1.0f : 0.0f;
          v3[r] = (vd > 1.0f) ? 0.0f : vd;
          i3[r] = id + cur;
          acc3[r] += z;
        }
      }
    }
    __syncthreads();
  }

  // Write acc[16,2]: C/D layout -> lanes 0,1 carry rows 0..7, lanes 16,17 carry rows 8..15
  if (w == 0) {
    #pragma unroll
    for (int r = 0; r < 8; ++r) {
      if (lane < 2)                out[(b0 + r) * 2 + lane]            = acc3[r];
      if (lane >= 16 && lane < 18) out[(b0 + 8 + r) * 2 + (lane - 16)] = acc3[r];
    }
  }
}

extern "C" void kernel_launch(void* const* d_in, const int* in_sizes, int n_in,
                              void* d_out, int out_size, void* d_ws, size_t ws_size,
                              hipStream_t stream) {
  const float* x  = (const float*)d_in[0];
  const float* w1 = (const float*)d_in[1];
  const float* b1 = (const float*)d_in[2];
  const float* w2 = (const float*)d_in[3];
  const float* b2 = (const float*)d_in[4];
  const float* w3 = (const float*)d_in[5];
  const float* b3 = (const float*)d_in[6];
  float* out = (float*)d_out;

  dim3 grid(2048 / 16);                 // 128 workgroups x 8 waves
  dim3 block(256);
  size_t smem = (34816 + 2 * 16384) * sizeof(float);   // 264 KB of 320 KB/WGP LDS
  snn_lif3_fused<<<grid, block, smem, stream>>>(x, w1, b1, w2, b2, w3, b3, out);
}